// GAT_24713241821309
// MI455X (gfx1250) — compile-verified
//
#include <hip/hip_runtime.h>
#include <math.h>

// ---- problem constants (match reference) ----
#define N_NODES   2000
#define F_IN      256
#define HID       64
#define HEADS     4
#define E_EDGES   16000
#define E_TOT     (E_EDGES + N_NODES)   // self loops appended: 18000
#define HC        (HEADS * HID)         // 256  (layer-1 output width)
#define K2DIM     (HEADS * HC)          // 1024 (layer-2 GEMM K: heads concatenated)
#define W2_COLS   (HEADS * N_NODES)     // 8000
#define MPAD      2048                  // padded row count for 64-row WMMA strips
#define NEG_SLOPE 0.2f

typedef float v2f __attribute__((ext_vector_type(2)));
typedef float v8f __attribute__((ext_vector_type(8)));

// ---- helpers ----
// monotone float <-> uint encoding so float max == unsigned max
__device__ __forceinline__ unsigned fenc(float f) {
  unsigned u = __float_as_uint(f);
  return (u & 0x80000000u) ? ~u : (u | 0x80000000u);
}
__device__ __forceinline__ float fdec(unsigned e) {
  return __uint_as_float((e & 0x80000000u) ? (e & 0x7FFFFFFFu) : ~e);
}
__device__ __forceinline__ float leakyrelu(float x) { return x > 0.f ? x : NEG_SLOPE * x; }
__device__ __forceinline__ void edge_sd(const int* __restrict__ ei, int e, int& s, int& d) {
  if (e < E_EDGES) { s = ei[e]; d = ei[E_EDGES + e]; }
  else             { s = e - E_EDGES; d = s; }   // self loop
}

// ---- zero scratch accumulators ----
__global__ void k_zero(float* __restrict__ p, int n) {
  int i = blockIdx.x * blockDim.x + threadIdx.x;
  if (i < n) p[i] = 0.f;
}

// ---- layer-1 GEMM: H1[2000,256] = X[2000,256] @ W1[256,256], fp32 WMMA ----
// One wave per block; wave owns a 64x32 strip: 4 M-tiles x 2 N-tiles,
// A-frag reused across both N-tiles -> 8 loads : 8 WMMAs per k-step.
__global__ __launch_bounds__(32)
void k_gemm1(const float* __restrict__ X, const float* __restrict__ W1,
             float* __restrict__ H1) {
  const int lane = threadIdx.x & 31;
  const int l15  = lane & 15;
  const int kh   = (lane >> 4) * 2;          // lane half selects K{0,1} vs K{2,3}
  const int na   = blockIdx.x * 32 + l15;    // N = 256 = 8 blocks * 32, always in range
  const int nb   = na + 16;
  const int m0   = blockIdx.y * 64;

  v8f acc[8] = {};                            // acc[t*2+j], t = M-tile, j = N-tile
  int row[4];
#pragma unroll
  for (int t = 0; t < 4; ++t) {
    int r = m0 + t * 16 + l15;
    row[t] = r < N_NODES ? r : (N_NODES - 1); // clamp reads; stores guarded
  }
  for (int k0 = 0; k0 < F_IN; k0 += 4) {
    const int k = k0 + kh;
    v2f bA, bB;
    bA.x = W1[(k    ) * HC + na];
    bA.y = W1[(k + 1) * HC + na];
    bB.x = W1[(k    ) * HC + nb];
    bB.y = W1[(k + 1) * HC + nb];
#pragma unroll
    for (int t = 0; t < 4; ++t) {
      v2f a;
      a.x = X[row[t] * F_IN + k];
      a.y = X[row[t] * F_IN + k + 1];
      acc[t * 2 + 0] = __builtin_amdgcn_wmma_f32_16x16x4_f32(false, a, false, bA,
                                            (short)0, acc[t * 2 + 0], false, false);
      acc[t * 2 + 1] = __builtin_amdgcn_wmma_f32_16x16x4_f32(false, a, false, bB,
                                            (short)0, acc[t * 2 + 1], false, false);
    }
  }
  const bool full = (m0 + 64 <= N_NODES);     // 31/32 blocks skip per-row guards
#pragma unroll
  for (int t = 0; t < 4; ++t)
#pragma unroll
    for (int r = 0; r < 8; ++r) {
      int mr = m0 + t * 16 + (lane >> 4) * 8 + r;  // C/D layout: lane-half => M/M+8
      if (full || mr < N_NODES) {
        H1[mr * HC + na] = acc[t * 2 + 0][r];
        H1[mr * HC + nb] = acc[t * 2 + 1][r];
      }
    }
}

// ---- layer-1 attention logits: a[n,h] = <H1[n,h,:], att[h,:]> ----
__global__ void k_att1(const float* __restrict__ H1, const float* __restrict__ as,
                       const float* __restrict__ ad, float* __restrict__ As,
                       float* __restrict__ Ad) {
  int i = blockIdx.x * blockDim.x + threadIdx.x;
  if (i >= N_NODES * HEADS) return;
  int n = i >> 2, h = i & 3;
  const float* hp = H1 + n * HC + h * HID;
  const float* sp = as + h * HID;
  const float* dp = ad + h * HID;
  float s0 = 0.f, s1 = 0.f;
  for (int c = 0; c < HID; ++c) { float v = hp[c]; s0 += v * sp[c]; s1 += v * dp[c]; }
  As[i] = s0; Ad[i] = s1;
}

// ---- segment max / sum over incoming edges (shared by both layers) ----
__global__ void k_edge_max(const int* __restrict__ ei, const float* __restrict__ As,
                           const float* __restrict__ Ad, unsigned* __restrict__ Menc) {
  int t = blockIdx.x * blockDim.x + threadIdx.x;
  if (t >= E_TOT * HEADS) return;
  int e = t >> 2, h = t & 3;
  int s, d; edge_sd(ei, e, s, d);
  float v = leakyrelu(As[s * HEADS + h] + Ad[d * HEADS + h]);
  atomicMax(&Menc[d * HEADS + h], fenc(v));
}
__global__ void k_edge_sum(const int* __restrict__ ei, const float* __restrict__ As,
                           const float* __restrict__ Ad, const unsigned* __restrict__ Menc,
                           float* __restrict__ S) {
  int t = blockIdx.x * blockDim.x + threadIdx.x;
  if (t >= E_TOT * HEADS) return;
  int e = t >> 2, h = t & 3;
  int s, d; edge_sd(ei, e, s, d);
  float v = leakyrelu(As[s * HEADS + h] + Ad[d * HEADS + h]);
  atomicAdd(&S[d * HEADS + h], expf(v - fdec(Menc[d * HEADS + h])));
}

// ---- layer-1 weighted scatter: AGG[dst,:] += alpha * H1[src,:]  (256 f32/edge) ----
__global__ __launch_bounds__(256)
void k_edge_agg1(const int* __restrict__ ei, const float* __restrict__ As,
                 const float* __restrict__ Ad, const unsigned* __restrict__ Menc,
                 const float* __restrict__ S, const float* __restrict__ H1,
                 float* __restrict__ AGG) {
  const int e = blockIdx.x, t = threadIdx.x;
  int s, d; edge_sd(ei, e, s, d);
  const int h = t >> 6;                      // 64 channels per head
  float v = leakyrelu(As[s * HEADS + h] + Ad[d * HEADS + h]);
  float alpha = expf(v - fdec(Menc[d * HEADS + h])) / S[d * HEADS + h];
  atomicAdd(&AGG[d * HC + t], alpha * H1[s * HC + t]);
}

// ---- bias + ELU in place ----
__global__ void k_elu_bias(float* __restrict__ HACT, const float* __restrict__ b1) {
  int i = blockIdx.x * blockDim.x + threadIdx.x;
  if (i >= N_NODES * HC) return;
  float v = HACT[i] + b1[i & (HC - 1)];
  HACT[i] = v > 0.f ? v : expm1f(v);
}

// ---- fold layer-2 attention vectors through W2: w_att[h,k] = sum_c W2[k,h*N+c]*att2[h,c] ----
__global__ void k_watt2(const float* __restrict__ W2, const float* __restrict__ att_s,
                        const float* __restrict__ att_d, float* __restrict__ wsrc,
                        float* __restrict__ wdst) {
  int i = blockIdx.x * blockDim.x + threadIdx.x;   // i = h*HC + k
  if (i >= HEADS * HC) return;
  int h = i / HC, k = i % HC;
  const float* wrow = W2 + (size_t)k * W2_COLS + h * N_NODES;
  const float* sa = att_s + h * N_NODES;
  const float* da = att_d + h * N_NODES;
  float s0 = 0.f, s1 = 0.f;
  for (int c = 0; c < N_NODES; ++c) { float w = wrow[c]; s0 += w * sa[c]; s1 += w * da[c]; }
  wsrc[i] = s0; wdst[i] = s1;
}

// ---- layer-2 attention logits via folded vectors: a2[n,h] = <HACT[n,:], w_att[h,:]> ----
__global__ void k_att2(const float* __restrict__ HACT, const float* __restrict__ WS,
                       const float* __restrict__ WD, float* __restrict__ As,
                       float* __restrict__ Ad) {
  int i = blockIdx.x * blockDim.x + threadIdx.x;
  if (i >= N_NODES * HEADS) return;
  int n = i >> 2, h = i & 3;
  const float* hp = HACT + n * HC;
  const float* sp = WS + h * HC;
  const float* dp = WD + h * HC;
  float s0 = 0.f, s1 = 0.f;
  for (int k = 0; k < HC; ++k) { float v = hp[k]; s0 += v * sp[k]; s1 += v * dp[k]; }
  As[i] = s0; Ad[i] = s1;
}

// ---- layer-2 scatter: P[dst, h*256+k] += alpha_h * HACT[src,k] ----
__global__ __launch_bounds__(256)
void k_edge_agg2(const int* __restrict__ ei, const float* __restrict__ As,
                 const float* __restrict__ Ad, const unsigned* __restrict__ Menc,
                 const float* __restrict__ S, const float* __restrict__ HACT,
                 float* __restrict__ P) {
  const int e = blockIdx.x, t = threadIdx.x;
  int s, d; edge_sd(ei, e, s, d);
  float hv = HACT[s * HC + t];
#pragma unroll
  for (int h = 0; h < HEADS; ++h) {
    float v = leakyrelu(As[s * HEADS + h] + Ad[d * HEADS + h]);
    float alpha = expf(v - fdec(Menc[d * HEADS + h])) / S[d * HEADS + h];
    atomicAdd(&P[d * K2DIM + h * HC + t], alpha * hv);
  }
}

// ---- final GEMM: OUT[m,c] = 0.25 * sum_k P[m,k] * W2[(k&255), (k>>8)*N + c] + b2[c] ----
// B gathered from W2 in-flight (head-concat layout); P padded+zeroed to 2048 rows.
// 64x32 strip per wave; N grid = 63 blocks of 32 (second N-subtile of last block is
// clamp-read / guard-stored so EXEC stays all-ones through the WMMA loop).
__global__ __launch_bounds__(32)
void k_gemm3(const float* __restrict__ P, const float* __restrict__ W2,
             const float* __restrict__ b2, float* __restrict__ OUT) {
  const int lane = threadIdx.x & 31;
  const int l15  = lane & 15;
  const int kh   = (lane >> 4) * 2;
  const int na   = blockIdx.x * 32 + l15;           // always < 2000 (last block: <=1999)
  const int nbR  = na + 16;                         // may reach 2015 in last block
  const int nb   = nbR < N_NODES ? nbR : (N_NODES - 1);
  const int m0   = blockIdx.y * 64;

  v8f acc[8] = {};
  for (int k0 = 0; k0 < K2DIM; k0 += 4) {
    const int k  = k0 + kh;
    const int k1 = k + 1;
    const size_t rowk  = (size_t)(k  & (HC - 1)) * W2_COLS + (k  >> 8) * N_NODES;
    const size_t rowk1 = (size_t)(k1 & (HC - 1)) * W2_COLS + (k1 >> 8) * N_NODES;
    v2f bA, bB;
    bA.x = W2[rowk + na];  bA.y = W2[rowk1 + na];
    bB.x = W2[rowk + nb];  bB.y = W2[rowk1 + nb];
#pragma unroll
    for (int t = 0; t < 4; ++t) {
      const int r = m0 + t * 16 + l15;              // P rows padded to MPAD, no clamp
      v2f a;
      a.x = P[r * K2DIM + k];
      a.y = P[r * K2DIM + k + 1];
      acc[t * 2 + 0] = __builtin_amdgcn_wmma_f32_16x16x4_f32(false, a, false, bA,
                                            (short)0, acc[t * 2 + 0], false, false);
      acc[t * 2 + 1] = __builtin_amdgcn_wmma_f32_16x16x4_f32(false, a, false, bB,
                                            (short)0, acc[t * 2 + 1], false, false);
    }
  }
  const float biasA = b2[na];
  const float biasB = b2[nb];
  const bool fullM = (m0 + 64 <= N_NODES);
  const bool okB   = (nbR < N_NODES);
#pragma unroll
  for (int t = 0; t < 4; ++t)
#pragma unroll
    for (int r = 0; r < 8; ++r) {
      int mr = m0 + t * 16 + (lane >> 4) * 8 + r;
      if (fullM || mr < N_NODES) {
        OUT[mr * N_NODES + na] = 0.25f * acc[t * 2 + 0][r] + biasA;
        if (okB) OUT[mr * N_NODES + nbR] = 0.25f * acc[t * 2 + 1][r] + biasB;
      }
    }
}

extern "C" void kernel_launch(void* const* d_in, const int* in_sizes, int n_in,
                              void* d_out, int out_size, void* d_ws, size_t ws_size,
                              hipStream_t stream) {
  (void)in_sizes; (void)n_in; (void)out_size; (void)ws_size;
  const float* X   = (const float*)d_in[0];
  const int*   EI  = (const int*)  d_in[1];
  const float* W1  = (const float*)d_in[2];
  const float* AS1 = (const float*)d_in[3];
  const float* AD1 = (const float*)d_in[4];
  const float* B1  = (const float*)d_in[5];
  const float* W2  = (const float*)d_in[6];
  const float* AS2 = (const float*)d_in[7];
  const float* AD2 = (const float*)d_in[8];
  const float* B2  = (const float*)d_in[9];
  float* OUT = (float*)d_out;

  // workspace layout: zeroed accumulators first, then read-write temps (~12.9 MB total)
  float* w = (float*)d_ws;
  float*    HACT = w;            w += MPAD * HC;      // layer-1 aggregate -> ELU activation
  float*    Pm   = w;            w += MPAD * K2DIM;   // layer-2 per-head aggregate (concat)
  unsigned* M1   = (unsigned*)w; w += MPAD * HEADS;
  float*    S1   = w;            w += MPAD * HEADS;
  unsigned* M2   = (unsigned*)w; w += MPAD * HEADS;
  float*    S2   = w;            w += MPAD * HEADS;
  const int ZN = MPAD * HC + MPAD * K2DIM + 4 * MPAD * HEADS;  // elements to zero
  float* H1   = w; w += MPAD * HC;
  float* Asr1 = w; w += MPAD * HEADS;
  float* Ads1 = w; w += MPAD * HEADS;
  float* Asr2 = w; w += MPAD * HEADS;
  float* Ads2 = w; w += MPAD * HEADS;
  float* WS2  = w; w += HEADS * HC;
  float* WD2  = w; w += HEADS * HC;

  k_zero<<<(ZN + 255) / 256, 256, 0, stream>>>((float*)d_ws, ZN);

  // layer 1
  k_gemm1<<<dim3(HC / 32, MPAD / 64), 32, 0, stream>>>(X, W1, H1);
  k_att1<<<(N_NODES * HEADS + 255) / 256, 256, 0, stream>>>(H1, AS1, AD1, Asr1, Ads1);
  k_edge_max<<<(E_TOT * HEADS + 255) / 256, 256, 0, stream>>>(EI, Asr1, Ads1, M1);
  k_edge_sum<<<(E_TOT * HEADS + 255) / 256, 256, 0, stream>>>(EI, Asr1, Ads1, M1, S1);
  k_edge_agg1<<<E_TOT, 256, 0, stream>>>(EI, Asr1, Ads1, M1, S1, H1, HACT);
  k_elu_bias<<<(N_NODES * HC + 255) / 256, 256, 0, stream>>>(HACT, B1);

  // layer 2 (attention via folded W2@att vectors; h2 never materialized)
  k_watt2<<<(HEADS * HC + 255) / 256, 256, 0, stream>>>(W2, AS2, AD2, WS2, WD2);
  k_att2<<<(N_NODES * HEADS + 255) / 256, 256, 0, stream>>>(HACT, WS2, WD2, Asr2, Ads2);
  k_edge_max<<<(E_TOT * HEADS + 255) / 256, 256, 0, stream>>>(EI, Asr2, Ads2, M2);
  k_edge_sum<<<(E_TOT * HEADS + 255) / 256, 256, 0, stream>>>(EI, Asr2, Ads2, M2, S2);
  k_edge_agg2<<<E_TOT, 256, 0, stream>>>(EI, Asr2, Ads2, M2, S2, HACT, Pm);
  k_gemm3<<<dim3((N_NODES + 31) / 32, MPAD / 64), 32, 0, stream>>>(Pm, W2, B2, OUT);
}